// BHGNN_25778393710881
// MI455X (gfx1250) — compile-verified
//
// BHGNN on MI455X (gfx1250). Compile-only target.
//
// Roofline: the only real GEMM is x@W_inner (2.15 GFLOP f32); the pipeline is
// HBM-bound (~100-150MB traffic -> ~5-7us @ 23.3TB/s), so we keep full fp32
// and use V_WMMA_F32_16X16X4_F32 (exact) rather than bf16/f16 WMMA.
// Edge gather (512MB logical) is served from LDS (graph tile = 64KB).
// Workspace use: ~42.3 MB of d_ws.

#include <hip/hip_runtime.h>
#include <hip/hip_bf16.h>

typedef float v2f __attribute__((ext_vector_type(2)));
typedef float v8f __attribute__((ext_vector_type(8)));

#define NEG_INF (-3.402823466e38f)

namespace {
constexpr int NN    = 65536;   // nodes
constexpr int KE    = 16;      // out-degree
constexpr int EE    = NN * KE; // edges
constexpr int FIN   = 128;
constexpr int NH    = 128;
constexpr int NB    = 512;     // graphs
constexpr int HH    = 6;       // hypernodes per graph
constexpr int NHYP  = NB * HH; // 3072

// workspace layout (floats)
constexpr size_t WS_XL    = 0;                    // [NN]
constexpr size_t WS_XR    = WS_XL + NN;           // [NN]
constexpr size_t WS_ATTR  = WS_XR + NN;           // [EE]
constexpr size_t WS_DINV  = WS_ATTR + EE;         // [NN]
constexpr size_t WS_XW    = WS_DINV + NN;         // [NN*NH]
constexpr size_t WS_XH    = WS_XW + (size_t)NN*NH;// [NHYP*256]
constexpr size_t WS_WL    = WS_XH + (size_t)NHYP*256; // [NHYP]
constexpr size_t WS_WR    = WS_WL + NHYP;         // [NHYP]
constexpr size_t WS_ATTRH = WS_WR + NHYP;         // [NHYP*HH]
constexpr size_t WS_DINVH = WS_ATTRH + (size_t)NHYP*HH; // [NHYP]
constexpr size_t WS_X1    = WS_DINVH + NHYP;      // [NB*256]
} // namespace

// ---------------------------------------------------------------------------
// Kernel A: xl = x @ att[:128], xr = x @ att[128:256].  One wave per node,
// coalesced float reads, xor-shuffle reduction (wave32).
// ---------------------------------------------------------------------------
__global__ void k_att_inner(const float* __restrict__ x,
                            const float* __restrict__ att,
                            float* __restrict__ xl, float* __restrict__ xr) {
  int wid = threadIdx.x >> 5, lane = threadIdx.x & 31;
  int node = blockIdx.x * 8 + wid;
  const float* xb = x + node * FIN;
  float pl = 0.f, pr = 0.f;
  #pragma unroll
  for (int j = lane; j < FIN; j += 32) {
    float v = xb[j];
    pl += v * att[j];
    pr += v * att[FIN + j];
  }
  #pragma unroll
  for (int off = 16; off; off >>= 1) {
    pl += __shfl_xor(pl, off);
    pr += __shfl_xor(pr, off);
  }
  if (lane == 0) { xl[node] = pl; xr[node] = pr; }
}

// ---------------------------------------------------------------------------
// Kernel B: per-node edge weights (relu), sparsemax over K=16, dinv.
// row = repeat(arange(N),16) -> node i owns edges [16i,16i+16).
// ---------------------------------------------------------------------------
__global__ void k_sparsemax(const float* __restrict__ xl,
                            const float* __restrict__ xr,
                            const int* __restrict__ col,
                            float* __restrict__ attr,
                            float* __restrict__ dinv) {
  int node = blockIdx.x * blockDim.x + threadIdx.x;
  if (node >= NN) return;
  float xli = xl[node];
  const int* cb = col + node * KE;
  float w[KE], s[KE];
  #pragma unroll
  for (int e = 0; e < KE; ++e) {
    float t = xli + xr[cb[e]];
    w[e] = fmaxf(t, 0.f);            // leaky_relu slope 0 == relu
    s[e] = w[e];
  }
  // sort descending (insertion sort, 16 elems)
  for (int i = 1; i < KE; ++i) {
    float key = s[i]; int j = i - 1;
    while (j >= 0 && s[j] < key) { s[j + 1] = s[j]; --j; }
    s[j + 1] = key;
  }
  float cum = 0.f, cumsel = 0.f; int kz = 1;
  #pragma unroll
  for (int i = 0; i < KE; ++i) {
    cum += s[i];
    if (1.0f + (float)(i + 1) * s[i] > cum) { kz = i + 1; cumsel = cum; }
  }
  float tau = (cumsel - 1.0f) / (float)kz;
  float ssum = 0.f;
  #pragma unroll
  for (int e = 0; e < KE; ++e) {
    float a = fmaxf(w[e] - tau, 0.f);
    attr[node * KE + e] = a;
    ssum += a;
  }
  dinv[node] = rsqrtf(1.0f + ssum);
}

// ---------------------------------------------------------------------------
// Kernel C: xw = x @ W_inner  (65536x128 @ 128x128) via V_WMMA_F32_16X16X4.
// W staged transposed in LDS, row stride 132 dwords (bank-conflict-free,
// 8B-aligned v2f reads). Block = 4 waves, each wave: 16 rows x 128 cols.
// f32 A 16x4 layout (ISA 7.12.2): lane m=l&15, kh=l>>4; a[j]=A[m][k0+2kh+j].
// B 4x16 mirrored: b[j]=B[k0+2kh+j][n].  C/D: row r+8kh, col n.
// ---------------------------------------------------------------------------
__global__ void k_gemm_wmma(const float* __restrict__ x,
                            const float* __restrict__ W,
                            float* __restrict__ xw) {
  extern __shared__ float WT[]; // [128 cols][stride 132]
  for (int idx = threadIdx.x; idx < FIN * NH; idx += blockDim.x) {
    int k = idx >> 7, n = idx & 127;
    WT[n * 132 + k] = W[idx];
  }
  __syncthreads();
  int wid = threadIdx.x >> 5, lane = threadIdx.x & 31;
  int kh = lane >> 4, n = lane & 15, m = lane & 15;
  int m_base = (blockIdx.x * 4 + wid) * 16;

#if __has_builtin(__builtin_amdgcn_wmma_f32_16x16x4_f32)
  v8f acc[8] = {};
  const float* xrow = x + (m_base + m) * FIN + 2 * kh;
  for (int k0 = 0; k0 < FIN; k0 += 4) {
    v2f a = *reinterpret_cast<const v2f*>(xrow + k0);
    #pragma unroll
    for (int nt = 0; nt < 8; ++nt) {
      v2f b = *reinterpret_cast<const v2f*>(&WT[(nt * 16 + n) * 132 + k0 + 2 * kh]);
      acc[nt] = __builtin_amdgcn_wmma_f32_16x16x4_f32(
          false, a, false, b, (short)0, acc[nt], false, false);
    }
  }
  #pragma unroll
  for (int nt = 0; nt < 8; ++nt)
    #pragma unroll
    for (int r = 0; r < 8; ++r)
      xw[(m_base + r + 8 * kh) * NH + nt * 16 + n] = acc[nt][r];
#else
  // scalar fallback (keeps the file compiling if builtin is absent)
  for (int idx = threadIdx.x; idx < 64 * NH; idx += blockDim.x) {
    int r = idx >> 7, c = idx & 127;
    int row = (blockIdx.x * 64) + r;
    float acc = 0.f;
    const float* xr_ = x + row * FIN;
    for (int k = 0; k < FIN; ++k) acc += xr_[k] * WT[c * 132 + k];
    xw[row * NH + c] = acc;
  }
  (void)kh; (void)n; (void)m; (void)m_base;
#endif
}

// ---------------------------------------------------------------------------
// Kernel D: per-graph GCN aggregate + bias + relu, fused with hypernode
// mean/max pooling (h never hits HBM).  One block per 128-node graph.
// Dynamic LDS: xw tile 64KB + norm 8KB + col 8KB + dinv 0.5KB = 82,432 B
// (legal: CDNA5 allows up to 320KB LDS per workgroup).
// Thread (f = tid&127, p = tid>>7) owns feature f of nodes i = 2t+p.
// Residues mod 6 have parity p, so pooling stats live entirely in 3 regs.
// ---------------------------------------------------------------------------
__global__ void k_agg_pool(const float* __restrict__ xw,
                           const float* __restrict__ attr,
                           const float* __restrict__ dinv,
                           const int* __restrict__ col,
                           const float* __restrict__ bin,
                           float* __restrict__ xh) {
  extern __shared__ float sm[];
  float* xwL   = sm;                       // 16384 f32
  float* normL = xwL + 16384;              // 2048 f32
  int*   colL  = (int*)(normL + 2048);     // 2048 i32
  float* dinvL = (float*)(colL + 2048);    // 128 f32

  int g = blockIdx.x, tid = threadIdx.x;
  const float4* src = (const float4*)(xw + (size_t)g * 16384);
  float4* dst4 = (float4*)xwL;
  for (int idx = tid; idx < 4096; idx += 256) dst4[idx] = src[idx];
  for (int idx = tid; idx < 2048; idx += 256) {
    normL[idx] = attr[g * 2048 + idx];
    colL[idx]  = col[g * 2048 + idx] - g * 128;   // graph-local
  }
  if (tid < 128) dinvL[tid] = dinv[g * 128 + tid];
  __syncthreads();
  for (int idx = tid; idx < 2048; idx += 256)
    normL[idx] *= dinvL[idx >> 4] * dinvL[colL[idx]];
  __syncthreads();

  int p = tid >> 7, f = tid & 127;
  float bf = bin[f];
  float sum3[3] = {0.f, 0.f, 0.f};
  float max3[3] = {NEG_INF, NEG_INF, NEG_INF};
  int j = 0;
  for (int t = 0; t < 64; ++t) {
    int i = 2 * t + p;
    const float* nb = normL + i * KE;
    const int*   cb = colL + i * KE;
    float acc = 0.f;
    #pragma unroll
    for (int e = 0; e < KE; ++e) acc += nb[e] * xwL[cb[e] * NH + f];
    acc += dinvL[i] * dinvL[i] * xwL[i * NH + f];
    float h = fmaxf(acc + bf, 0.f);
    sum3[j] += h;
    max3[j] = fmaxf(max3[j], h);
    j = (j == 2) ? 0 : j + 1;
  }
  #pragma unroll
  for (int q = 0; q < 3; ++q) {
    int r = (p + 2 * q) % 6;           // residue mod 6 (parity == p)
    int cnt = (127 - r) / 6 + 1;       // 22 for r<2, else 21
    int hyp = g * HH + r;
    xh[hyp * 256 + f]       = sum3[q] / (float)cnt; // mean half
    xh[hyp * 256 + 128 + f] = max3[q];              // max half
  }
}

// ---------------------------------------------------------------------------
// Kernel E1: wl = xh @ att_outer[:256], wr = xh @ att_outer[256:512]
// ---------------------------------------------------------------------------
__global__ void k_att_outer(const float* __restrict__ xh,
                            const float* __restrict__ att,
                            float* __restrict__ wl, float* __restrict__ wr) {
  int wid = threadIdx.x >> 5, lane = threadIdx.x & 31;
  int node = blockIdx.x * 8 + wid;
  const float* xb = xh + node * 256;
  float pl = 0.f, pr = 0.f;
  #pragma unroll
  for (int j = lane; j < 256; j += 32) {
    float v = xb[j];
    pl += v * att[j];
    pr += v * att[256 + j];
  }
  #pragma unroll
  for (int off = 16; off; off >>= 1) {
    pl += __shfl_xor(pl, off);
    pr += __shfl_xor(pr, off);
  }
  if (lane == 0) { wl[node] = pl; wr[node] = pr; }
}

// ---------------------------------------------------------------------------
// Kernel E2: leaky_relu(0.2) + softmax over each hypernode's 6-clique; dinvh.
// ---------------------------------------------------------------------------
__global__ void k_softmax_outer(const float* __restrict__ wl,
                                const float* __restrict__ wr,
                                float* __restrict__ attrh,
                                float* __restrict__ dinvh) {
  int r = blockIdx.x * blockDim.x + threadIdx.x;
  if (r >= NHYP) return;
  int g = r / HH;
  float wli = wl[r];
  float wv[HH], mx = NEG_INF;
  #pragma unroll
  for (int j = 0; j < HH; ++j) {
    float t = wli + wr[g * HH + j];
    t = (t > 0.f) ? t : 0.2f * t;
    wv[j] = t;
    mx = fmaxf(mx, t);
  }
  float se = 0.f;
  #pragma unroll
  for (int j = 0; j < HH; ++j) { wv[j] = expf(wv[j] - mx); se += wv[j]; }
  float inv = 1.0f / se, ssum = 0.f;
  #pragma unroll
  for (int j = 0; j < HH; ++j) {
    float a = wv[j] * inv;
    attrh[r * HH + j] = a;
    ssum += a;
  }
  dinvh[r] = rsqrtf(1.0f + ssum);
}

// ---------------------------------------------------------------------------
// Kernel F: both hyper GCN layers + graph readout, fused per graph.
// Block = 128 threads (thread = output feature). 6 hypernodes live in LDS.
// ---------------------------------------------------------------------------
__global__ void k_hyper_gcn(const float* __restrict__ xh,
                            const float* __restrict__ attrh,
                            const float* __restrict__ dinvh,
                            const float* __restrict__ W1, const float* __restrict__ b1,
                            const float* __restrict__ W2, const float* __restrict__ b2,
                            float* __restrict__ x1) {
  __shared__ float xhL[HH * 256];
  __shared__ float z1L[HH * NH];
  __shared__ float attrL[HH * HH];
  __shared__ float dnvL[HH];
  int g = blockIdx.x, f = threadIdx.x;
  for (int idx = f; idx < HH * 256; idx += 128) xhL[idx] = xh[g * HH * 256 + idx];
  if (f < HH * HH) attrL[f] = attrh[g * HH * HH + f];
  if (f < HH) dnvL[f] = dinvh[g * HH + f];
  __syncthreads();

  // layer 1: 256 -> 128
  float xw1[HH] = {0.f, 0.f, 0.f, 0.f, 0.f, 0.f};
  for (int k = 0; k < 256; ++k) {
    float wv = W1[k * NH + f];
    #pragma unroll
    for (int jn = 0; jn < HH; ++jn) xw1[jn] += xhL[jn * 256 + k] * wv;
  }
  float bb1 = b1[f];
  #pragma unroll
  for (int i = 0; i < HH; ++i) {
    float acc = 0.f;
    #pragma unroll
    for (int jn = 0; jn < HH; ++jn)
      acc += dnvL[i] * attrL[i * HH + jn] * dnvL[jn] * xw1[jn];
    acc += dnvL[i] * dnvL[i] * xw1[i];
    z1L[i * NH + f] = fmaxf(acc + bb1, 0.f);
  }
  __syncthreads();

  // layer 2: 128 -> 128
  float xw2[HH] = {0.f, 0.f, 0.f, 0.f, 0.f, 0.f};
  for (int k = 0; k < NH; ++k) {
    float wv = W2[k * NH + f];
    #pragma unroll
    for (int jn = 0; jn < HH; ++jn) xw2[jn] += z1L[jn * NH + k] * wv;
  }
  float bb2 = b2[f];
  float z2[HH];
  #pragma unroll
  for (int i = 0; i < HH; ++i) {
    float acc = 0.f;
    #pragma unroll
    for (int jn = 0; jn < HH; ++jn)
      acc += dnvL[i] * attrL[i * HH + jn] * dnvL[jn] * xw2[jn];
    acc += dnvL[i] * dnvL[i] * xw2[i];
    z2[i] = fmaxf(acc + bb2, 0.f);
  }

  // readout: mean || max over 6 hypernodes
  float sm_ = 0.f, mx = NEG_INF;
  #pragma unroll
  for (int i = 0; i < HH; ++i) { sm_ += z2[i]; mx = fmaxf(mx, z2[i]); }
  x1[g * 256 + f]       = sm_ * (1.0f / 6.0f);
  x1[g * 256 + 128 + f] = mx;
}

// ---------------------------------------------------------------------------
// Kernel H: MLP heads, fused per graph. 512 threads: tids 0-255 = m branch,
// 256-511 = v branch.  Outputs: out1[512,10] | out2[512,10] | m[512,128].
// ---------------------------------------------------------------------------
__global__ void k_mlp_heads(const float* __restrict__ x1,
                            const float* __restrict__ f1mW, const float* __restrict__ f1mb,
                            const float* __restrict__ f2mW, const float* __restrict__ f2mb,
                            const float* __restrict__ f1vW, const float* __restrict__ f1vb,
                            const float* __restrict__ f2vW, const float* __restrict__ f2vb,
                            const float* __restrict__ cmW, const float* __restrict__ cmb,
                            const float* __restrict__ cvW, const float* __restrict__ cvb,
                            float* __restrict__ out) {
  __shared__ float x1L[256];
  __shared__ float tm[256], tv[256];
  __shared__ float mm[128], vv[128];
  int g = blockIdx.x, tid = threadIdx.x;
  int br = tid >> 8, j = tid & 255;
  if (tid < 256) x1L[tid] = x1[g * 256 + tid];
  __syncthreads();

  const float* W1 = br ? f1vW : f1mW;
  const float* B1 = br ? f1vb : f1mb;
  float acc = B1[j];
  for (int k = 0; k < 256; ++k) acc += x1L[k] * W1[k * 256 + j];
  (br ? tv : tm)[j] = fmaxf(acc, 0.f);
  __syncthreads();

  if (j < 128) {
    const float* W2 = br ? f2vW : f2mW;
    const float* B2 = br ? f2vb : f2mb;
    const float* T  = br ? tv : tm;
    float a2 = B2[j];
    for (int k = 0; k < 256; ++k) a2 += T[k] * W2[k * 128 + j];
    float rr = fmaxf(a2, 0.f);
    (br ? vv : mm)[j] = rr;
    if (!br) out[10240 + g * 128 + j] = rr;   // third output: m
  }
  __syncthreads();

  if (j < 10) {
    const float* CW = br ? cvW : cmW;
    const float* CB = br ? cvb : cmb;
    const float* M  = br ? vv : mm;
    float a3 = CB[j];
    for (int k = 0; k < 128; ++k) a3 += M[k] * CW[k * 10 + j];
    out[br * 5120 + g * 10 + j] = a3;
  }
}

// ---------------------------------------------------------------------------
extern "C" void kernel_launch(void* const* d_in, const int* in_sizes, int n_in,
                              void* d_out, int out_size, void* d_ws, size_t ws_size,
                              hipStream_t stream) {
  const float* x        = (const float*)d_in[0];
  const int*   ei       = (const int*)d_in[1];     // [2,E]; col = ei + E
  const float* att_in   = (const float*)d_in[5];
  const float* W_inner  = (const float*)d_in[6];
  const float* b_inner  = (const float*)d_in[7];
  const float* att_out  = (const float*)d_in[8];
  const float* W_out1   = (const float*)d_in[9];
  const float* b_out1   = (const float*)d_in[10];
  const float* W_out2   = (const float*)d_in[11];
  const float* b_out2   = (const float*)d_in[12];
  const float* f1mW = (const float*)d_in[13]; const float* f1mb = (const float*)d_in[14];
  const float* f2mW = (const float*)d_in[15]; const float* f2mb = (const float*)d_in[16];
  const float* f1vW = (const float*)d_in[17]; const float* f1vb = (const float*)d_in[18];
  const float* f2vW = (const float*)d_in[19]; const float* f2vb = (const float*)d_in[20];
  const float* cmW  = (const float*)d_in[21]; const float* cmb  = (const float*)d_in[22];
  const float* cvW  = (const float*)d_in[23]; const float* cvb  = (const float*)d_in[24];
  float* out = (float*)d_out;

  float* ws    = (float*)d_ws;
  float* xl    = ws + WS_XL;
  float* xr    = ws + WS_XR;
  float* attr  = ws + WS_ATTR;
  float* dinv  = ws + WS_DINV;
  float* xw    = ws + WS_XW;
  float* xh    = ws + WS_XH;
  float* wl    = ws + WS_WL;
  float* wr    = ws + WS_WR;
  float* attrh = ws + WS_ATTRH;
  float* dinvh = ws + WS_DINVH;
  float* x1    = ws + WS_X1;

  const int* col = ei + EE;

  const size_t gemm_smem = 132 * 128 * sizeof(float);                 // 67,584 B
  const size_t agg_smem  = (16384 + 2048 + 2048 + 128) * sizeof(float); // 82,432 B
  (void)hipFuncSetAttribute((const void*)k_gemm_wmma,
                            hipFuncAttributeMaxDynamicSharedMemorySize, (int)gemm_smem);
  (void)hipFuncSetAttribute((const void*)k_agg_pool,
                            hipFuncAttributeMaxDynamicSharedMemorySize, (int)agg_smem);

  k_att_inner   <<<NN / 8,    256, 0, stream>>>(x, att_in, xl, xr);
  k_sparsemax   <<<NN / 256,  256, 0, stream>>>(xl, xr, col, attr, dinv);
  k_gemm_wmma   <<<NN / 64,   128, gemm_smem, stream>>>(x, W_inner, xw);
  k_agg_pool    <<<NB,        256, agg_smem, stream>>>(xw, attr, dinv, col, b_inner, xh);
  k_att_outer   <<<NHYP / 8,  256, 0, stream>>>(xh, att_out, wl, wr);
  k_softmax_outer<<<NHYP / 256, 256, 0, stream>>>(wl, wr, attrh, dinvh);
  k_hyper_gcn   <<<NB,        128, 0, stream>>>(xh, attrh, dinvh,
                                                W_out1, b_out1, W_out2, b_out2, x1);
  k_mlp_heads   <<<NB,        512, 0, stream>>>(x1, f1mW, f1mb, f2mW, f2mb,
                                                f1vW, f1vb, f2vW, f2vb,
                                                cmW, cmb, cvW, cvb, out);
  (void)in_sizes; (void)n_in; (void)out_size; (void)ws_size;
}